// ContextBlock2d_71545565217092
// MI455X (gfx1250) — compile-verified
//
#include <hip/hip_runtime.h>
#include <hip/hip_bf16.h>
#include <math.h>

// Problem constants (match reference setup_inputs)
#define BB 32
#define CC 512
#define CR 32
#define HWS 4096
#define NCHUNK 8              // channel chunks for logits partials
#define CPC (CC / NCHUNK)     // 64 channels per chunk
#define SCHUNK 1024           // spatial positions per block in logits kernel
#define EPSV 1e-5f

typedef __attribute__((ext_vector_type(2))) float v2f;
typedef __attribute__((ext_vector_type(8))) float v8f;

// ---------------------------------------------------------------------------
// K1: logits partials.  Reads x once (256 MB).  Each block: (s-chunk, c-chunk, b).
// Thread t owns one float4 of spatial positions; loops over 64 channels with
// in-register accumulation; w_mask chunk broadcast from LDS.
// ---------------------------------------------------------------------------
__global__ void k_logits_partial(const float* __restrict__ x,
                                 const float* __restrict__ w_mask,
                                 float* __restrict__ logits_part) {
    const int sblk = blockIdx.x, cblk = blockIdx.y, b = blockIdx.z;
    const int t = threadIdx.x;
    __shared__ float wm[CPC];
    if (t < CPC) wm[t] = w_mask[cblk * CPC + t];
    __syncthreads();

    const float* xp = x + ((size_t)b * CC + (size_t)cblk * CPC) * HWS
                        + (size_t)sblk * SCHUNK + (size_t)t * 4;
    float4 acc = {0.f, 0.f, 0.f, 0.f};
#pragma unroll 4
    for (int c = 0; c < CPC; ++c) {
        float4 v = *(const float4*)(xp + (size_t)c * HWS);
        float w = wm[c];
        acc.x = fmaf(v.x, w, acc.x);
        acc.y = fmaf(v.y, w, acc.y);
        acc.z = fmaf(v.z, w, acc.z);
        acc.w = fmaf(v.w, w, acc.w);
    }
    float4* out = (float4*)(logits_part + ((size_t)cblk * BB + b) * HWS
                            + (size_t)sblk * SCHUNK + (size_t)t * 4);
    *out = acc;
}

// ---------------------------------------------------------------------------
// K2: reduce partials + softmax over HW per batch.  One block per b.
// Fixed-order tree reductions (deterministic).
// ---------------------------------------------------------------------------
__global__ void k_softmax(const float* __restrict__ logits_part,
                          const float* __restrict__ b_mask,
                          float* __restrict__ attn) {
    const int b = blockIdx.x;
    const int t = threadIdx.x;
    const int wave = t >> 5, lane = t & 31;
    const float bm = b_mask[0];
    __shared__ float red[8];

    float l[16];
    float mx = -3.0e38f;
#pragma unroll
    for (int i = 0; i < 16; ++i) {
        int s = t + i * 256;
        float v = bm;
#pragma unroll
        for (int p = 0; p < NCHUNK; ++p)
            v += logits_part[((size_t)p * BB + b) * HWS + s];
        l[i] = v;
        mx = fmaxf(mx, v);
    }
#pragma unroll
    for (int o = 16; o >= 1; o >>= 1) mx = fmaxf(mx, __shfl_xor(mx, o, 32));
    if (lane == 0) red[wave] = mx;
    __syncthreads();
    float bmx = red[0];
#pragma unroll
    for (int w = 1; w < 8; ++w) bmx = fmaxf(bmx, red[w]);
    __syncthreads();

    float sum = 0.f;
#pragma unroll
    for (int i = 0; i < 16; ++i) {
        l[i] = expf(l[i] - bmx);
        sum += l[i];
    }
#pragma unroll
    for (int o = 16; o >= 1; o >>= 1) sum += __shfl_xor(sum, o, 32);
    if (lane == 0) red[wave] = sum;
    __syncthreads();
    float tot = 0.f;
#pragma unroll
    for (int w = 0; w < 8; ++w) tot += red[w];
    float inv = 1.f / tot;
#pragma unroll
    for (int i = 0; i < 16; ++i) {
        int s = t + i * 256;
        attn[(size_t)b * HWS + s] = l[i] * inv;
    }
}

// ---------------------------------------------------------------------------
// K3: fused context + instance-norm moments.  One block per (b,c) row:
// a single read of x yields Sum(x), Sum(x^2), Sum(attn*x).  attn row (16 KB)
// is L2-hot across the 512 c-blocks of a batch.
// ---------------------------------------------------------------------------
__global__ void k_context_moments(const float* __restrict__ x,
                                  const float* __restrict__ attn,
                                  float* __restrict__ ctx,
                                  float* __restrict__ meanp,
                                  float* __restrict__ rstdp) {
    const int bc = blockIdx.x;
    const int b = bc / CC;
    const int t = threadIdx.x;
    const int wave = t >> 5, lane = t & 31;
    __shared__ float red[24];

    const float* xp = x + (size_t)bc * HWS;
    const float* ap = attn + (size_t)b * HWS;
    float s1 = 0.f, s2 = 0.f, s3 = 0.f;
#pragma unroll
    for (int i = 0; i < 4; ++i) {
        int s = (t + i * 256) * 4;
        float4 v = *(const float4*)(xp + s);
        float4 a = *(const float4*)(ap + s);
        s1 += (v.x + v.y) + (v.z + v.w);
        s2 += (v.x * v.x + v.y * v.y) + (v.z * v.z + v.w * v.w);
        s3 += (v.x * a.x + v.y * a.y) + (v.z * a.z + v.w * a.w);
    }
#pragma unroll
    for (int o = 16; o >= 1; o >>= 1) {
        s1 += __shfl_xor(s1, o, 32);
        s2 += __shfl_xor(s2, o, 32);
        s3 += __shfl_xor(s3, o, 32);
    }
    if (lane == 0) { red[wave] = s1; red[8 + wave] = s2; red[16 + wave] = s3; }
    __syncthreads();
    if (t == 0) {
        float a1 = 0.f, a2 = 0.f, a3 = 0.f;
#pragma unroll
        for (int w = 0; w < 8; ++w) { a1 += red[w]; a2 += red[8 + w]; a3 += red[16 + w]; }
        float m = a1 * (1.f / HWS);
        float var = a2 * (1.f / HWS) - m * m;
        ctx[bc] = a3;
        meanp[bc] = m;
        rstdp[bc] = rsqrtf(var + EPSV);
    }
}

// ---------------------------------------------------------------------------
// K4: bottleneck MLP gate on WMMA (V_WMMA_F32_16X16X4_F32).  Single block,
// 256 threads = 8 waves.
//   GEMM1: t[32,32]   = ctx[32,512] @ w1^T   (4 tiles, 128 k-steps each)
//   LN(Cr=32) + ReLU in LDS
//   GEMM2: z[32,512]  = t[32,32]    @ w2^T   (64 tiles, 8 k-steps each)
//   gate = sigmoid(z + b2)
// ISA layouts (wave32): A 16x4: lanes<16 hold K0,K1; lanes>=16 hold K2,K3.
// B 4x16: per-lane column N=lane%16, K halves split at lane 16.
// C/D: 8 VGPRs, element (M = i + 8*(lane>=16), N = lane%16).
// ---------------------------------------------------------------------------
__global__ void k_gate(const float* __restrict__ ctx,
                       const float* __restrict__ w1,
                       const float* __restrict__ b1,
                       const float* __restrict__ ln_g,
                       const float* __restrict__ ln_b,
                       const float* __restrict__ w2,
                       const float* __restrict__ b2,
                       float* __restrict__ gate) {
    __shared__ float tls[CR][CR + 1];
    const int t = threadIdx.x;
    const int wave = t >> 5;
    const int lane = t & 31;
    const int half = lane >> 4;     // 0: K lo pair, 1: K hi pair
    const int lm = lane & 15;

    // ---- GEMM1: waves 0..3 each own one 16x16 tile of t ----
    if (wave < 4) {
        const int mi = wave >> 1;   // row tile (batch)
        const int ni = wave & 1;    // col tile (Cr)
        const int m = mi * 16 + lm; // A row for this lane
        const int n = ni * 16 + lm; // B col for this lane
        v8f acc = {};
        for (int k0 = 0; k0 < CC; k0 += 4) {
            int ka = k0 + 2 * half;
            v2f a, bv;
            a.x = ctx[(size_t)m * CC + ka];
            a.y = ctx[(size_t)m * CC + ka + 1];
            bv.x = w1[(size_t)n * CC + ka];       // B[k][n] = w1[n][k]
            bv.y = w1[(size_t)n * CC + ka + 1];
            acc = __builtin_amdgcn_wmma_f32_16x16x4_f32(
                false, a, false, bv, (short)0, acc, false, false);
        }
        float bias = b1[n];
#pragma unroll
        for (int i = 0; i < 8; ++i)
            tls[mi * 16 + i + 8 * half][n] = acc[i] + bias;
    }
    __syncthreads();

    // ---- LayerNorm over Cr + ReLU, one thread per batch row ----
    if (t < CR) {
        float mu = 0.f;
#pragma unroll
        for (int j = 0; j < CR; ++j) mu += tls[t][j];
        mu *= (1.f / CR);
        float var = 0.f;
#pragma unroll
        for (int j = 0; j < CR; ++j) { float d = tls[t][j] - mu; var += d * d; }
        var *= (1.f / CR);
        float inv = rsqrtf(var + EPSV);
#pragma unroll
        for (int j = 0; j < CR; ++j) {
            float v = (tls[t][j] - mu) * inv * ln_g[j] + ln_b[j];
            tls[t][j] = fmaxf(v, 0.f);
        }
    }
    __syncthreads();

    // ---- GEMM2: 2x32 = 64 tiles over N=512; 8 waves, 8 tiles each ----
    for (int tile = wave; tile < 64; tile += 8) {
        const int mi = tile & 1;
        const int ni = tile >> 1;
        const int m = mi * 16 + lm;
        const int n = ni * 16 + lm;      // output channel c
        v8f acc = {};
#pragma unroll
        for (int k0 = 0; k0 < CR; k0 += 4) {
            int ka = k0 + 2 * half;
            v2f a, bv;
            a.x = tls[m][ka];
            a.y = tls[m][ka + 1];
            bv.x = w2[(size_t)n * CR + ka];       // B[k][n] = w2[n][k]
            bv.y = w2[(size_t)n * CR + ka + 1];
            acc = __builtin_amdgcn_wmma_f32_16x16x4_f32(
                false, a, false, bv, (short)0, acc, false, false);
        }
        float bias = b2[n];
#pragma unroll
        for (int i = 0; i < 8; ++i) {
            int row = mi * 16 + i + 8 * half;     // batch index
            float z = acc[i] + bias;
            gate[(size_t)row * CC + n] = 1.f / (1.f + expf(-z));
        }
    }
}

// ---------------------------------------------------------------------------
// K5: final blend.  Streaming read of x + write of out (float4), with
// per-(b,c) scalars gate/mean/rstd L2-resident.
// ---------------------------------------------------------------------------
__global__ void k_output(const float* __restrict__ x,
                         const float* __restrict__ gate,
                         const float* __restrict__ meanp,
                         const float* __restrict__ rstdp,
                         float* __restrict__ out) {
    size_t i4 = (size_t)blockIdx.x * blockDim.x + threadIdx.x;  // float4 index
    int bc = (int)(i4 >> 10);                                    // HWS/4 = 1024
    float g = gate[bc];
    float m = meanp[bc];
    float r = rstdp[bc];
    float og = 1.f - g;
    float4 v = ((const float4*)x)[i4];
    float4 o;
    o.x = v.x * g + og * (v.x - m) * r;
    o.y = v.y * g + og * (v.y - m) * r;
    o.z = v.z * g + og * (v.z - m) * r;
    o.w = v.w * g + og * (v.w - m) * r;
    ((float4*)out)[i4] = o;
}

// ---------------------------------------------------------------------------
extern "C" void kernel_launch(void* const* d_in, const int* in_sizes, int n_in,
                              void* d_out, int out_size, void* d_ws, size_t ws_size,
                              hipStream_t stream) {
    const float* x      = (const float*)d_in[0];
    const float* w_mask = (const float*)d_in[1];
    const float* b_mask = (const float*)d_in[2];
    const float* w1     = (const float*)d_in[3];
    const float* b1     = (const float*)d_in[4];
    const float* ln_g   = (const float*)d_in[5];
    const float* ln_b   = (const float*)d_in[6];
    const float* w2     = (const float*)d_in[7];
    const float* b2     = (const float*)d_in[8];
    float* out = (float*)d_out;

    // Workspace carve-up (~4.75 MB)
    float* ws          = (float*)d_ws;
    float* logits_part = ws;                                  // NCHUNK*BB*HWS
    float* attn        = logits_part + (size_t)NCHUNK * BB * HWS;
    float* ctx         = attn + (size_t)BB * HWS;
    float* meanp       = ctx + (size_t)BB * CC;
    float* rstdp       = meanp + (size_t)BB * CC;
    float* gate        = rstdp + (size_t)BB * CC;

    dim3 g1(HWS / SCHUNK, NCHUNK, BB);                        // 4 x 8 x 32
    k_logits_partial<<<g1, 256, 0, stream>>>(x, w_mask, logits_part);

    k_softmax<<<BB, 256, 0, stream>>>(logits_part, b_mask, attn);

    k_context_moments<<<BB * CC, 256, 0, stream>>>(x, attn, ctx, meanp, rstdp);

    k_gate<<<1, 256, 0, stream>>>(ctx, w1, b1, ln_g, ln_b, w2, b2, gate);

    const size_t n4 = (size_t)BB * CC * HWS / 4;              // 16,777,216
    k_output<<<(unsigned)(n4 / 256), 256, 0, stream>>>(x, gate, meanp, rstdp, out);
}